// SimplifiedEvoformer_34187939676349
// MI455X (gfx1250) — compile-verified
//
#include <hip/hip_runtime.h>
#include <hip/hip_bf16.h>

typedef float v2f __attribute__((ext_vector_type(2)));
typedef float v4f __attribute__((ext_vector_type(4)));
typedef float v8f __attribute__((ext_vector_type(8)));

#define L 1024
#define ND 64
#define ED 32
#define LN_EPS 1e-5f
#define WAVES 8
#define TROW 36   // padded LDS row stride in floats (16B-aligned rows, conflict-free frags)

// ---------------------------------------------------------------------------
// Kernel A: ni[i][f] = (node[i] @ W_pair[0:64])[f]
//           nj[i][f] = (node[i] @ W_pair[64:128])[f]   (separate compact arrays)
// ---------------------------------------------------------------------------
__global__ void evo_ninj(const float* __restrict__ node,
                         const float* __restrict__ W_pair,
                         float* __restrict__ ni,
                         float* __restrict__ nj) {
  __shared__ float nrow[ND];
  const int i = blockIdx.x;
  const int t = threadIdx.x;            // 0..63
  nrow[t] = node[i * ND + t];
  __syncthreads();
  const int base = (t < ED) ? 0 : ND;   // Wi rows or Wj rows
  const int f = t & (ED - 1);
  float acc = 0.f;
#pragma unroll 8
  for (int k = 0; k < ND; ++k)
    acc = fmaf(nrow[k], W_pair[(base + k) * ED + f], acc);
  if (t < ED) ni[i * ED + f] = acc;
  else        nj[i * ED + f] = acc;
}

// ---------------------------------------------------------------------------
// Kernel B: per block i: for all j-tiles, pair_out = pair + ni + nj + pair@Wp + b,
// fused with the j-axis mean (pair_to_node). WMMA f32 16x16x4, K=32 -> 8 steps,
// N=32 -> 2 column tiles. Pair streams are NT (read-once / write-once) to keep
// the 192MB L2 free for the heavily re-read nj table (1024 blocks x 128KB).
// ---------------------------------------------------------------------------
__global__ void evo_pair(const float* __restrict__ pairf,
                         const float* __restrict__ W_pair,
                         const float* __restrict__ b_pair,
                         const float* __restrict__ ni,
                         const float* __restrict__ nj,
                         float* __restrict__ pair_out,
                         float* __restrict__ p2n) {
  __shared__ float tile[WAVES][16 * TROW];   // pair tile (then finished tile)
  __shared__ float njt[WAVES][16 * TROW];    // nj[j0..j0+15][0:32] tile
  __shared__ float cni[ED];                  // ni[i][f] + b_pair[f]
  __shared__ float psum[ED];

  const int i    = blockIdx.x;
  const int tid  = threadIdx.x;
  const int wave = tid >> 5;
  const int lane = tid & 31;
  const int hi   = lane >> 4;   // 0: lanes 0-15, 1: lanes 16-31
  const int m16  = lane & 15;
  const int kb0  = hi * 2;      // A/B fragment K base within a K=4 step

  if (tid < ED) {
    cni[tid]  = ni[i * ED + tid] + b_pair[tid];
    psum[tid] = 0.f;
  }
  __syncthreads();

  // Wp B-fragments in registers (loop-invariant). ISA B layout (4x16 f32):
  // lanes 0-15 hold (K=0,K=1) col=lane&15; lanes 16-31 hold (K=2,K=3).
  v2f bfrag[8][2];
#pragma unroll
  for (int s = 0; s < 8; ++s) {
    const int k = kb0 + 4 * s;
#pragma unroll
    for (int nt = 0; nt < 2; ++nt) {
      const int f = nt * 16 + m16;
      bfrag[s][nt].x = W_pair[(2 * ND + k    ) * ED + f];
      bfrag[s][nt].y = W_pair[(2 * ND + k + 1) * ED + f];
    }
  }

  float sum0 = 0.f, sum1 = 0.f;
  float* tl = tile[wave];
  float* nl = njt[wave];

  for (int jt = wave; jt < L / 16; jt += WAVES) {
    const int j0 = jt * 16;
    const float* gsrc = pairf + ((size_t)i * L + j0) * ED;   // 512 contiguous floats
    const float* gnj  = nj + (size_t)j0 * ED;                // 512 contiguous floats

    // Coalesced global->LDS: pair tile NT (read-once), nj tile RT (hot in L2).
#pragma unroll
    for (int q = 0; q < 4; ++q) {
      const int idx4 = lane + 32 * q;   // float4 index within tile
      const int row  = idx4 >> 3;       // 8 float4 per 32-float row
      const int c4   = idx4 & 7;
      const v4f vp = __builtin_nontemporal_load((const v4f*)gsrc + idx4);
      const v4f vn = ((const v4f*)gnj)[idx4];
      *(v4f*)&tl[row * TROW + c4 * 4] = vp;
      *(v4f*)&nl[row * TROW + c4 * 4] = vn;
    }
    // Same-wave DS ops are in-order; compiler inserts s_wait_dscnt before reuse.

    // A fragments from LDS (ISA 16x4 f32 layout: lanes 0-15 K=0,1; 16-31 K=2,3)
    // + 16 chained WMMAs.
    v8f acc0 = {};
    v8f acc1 = {};
    const float* arow = &tl[m16 * TROW + kb0];
#pragma unroll
    for (int s = 0; s < 8; ++s) {
      const v2f a = *(const v2f*)(arow + 4 * s);
      acc0 = __builtin_amdgcn_wmma_f32_16x16x4_f32(false, a, false, bfrag[s][0],
                                                   (short)0, acc0, false, false);
      acc1 = __builtin_amdgcn_wmma_f32_16x16x4_f32(false, a, false, bfrag[s][1],
                                                   (short)0, acc1, false, false);
    }

    // Epilogue: C/D layout: VGPR v -> row m = v + 8*hi, col f = m16 (+16 tile1).
    // Residual + ni + b (cni) + nj (LDS tile), write back into the same LDS
    // slots (same lane, DS in-order), accumulate the j-sums.
#pragma unroll
    for (int v = 0; v < 8; ++v) {
      const int m = v + 8 * hi;
      const float o0 = acc0[v] + tl[m * TROW + m16]
                     + cni[m16]      + nl[m * TROW + m16];
      const float o1 = acc1[v] + tl[m * TROW + 16 + m16]
                     + cni[16 + m16] + nl[m * TROW + 16 + m16];
      tl[m * TROW + m16]      = o0;
      tl[m * TROW + 16 + m16] = o1;
      sum0 += o0;
      sum1 += o1;
    }

    // Coalesced NT store of the finished tile: 4 b128 stores/lane, 512B/instr.
    float* gdst = pair_out + ((size_t)i * L + j0) * ED;
#pragma unroll
    for (int q = 0; q < 4; ++q) {
      const int idx4 = lane + 32 * q;
      const int row  = idx4 >> 3;
      const int c4   = idx4 & 7;
      const v4f v4 = *(const v4f*)&tl[row * TROW + c4 * 4];
      __builtin_nontemporal_store(v4, (v4f*)gdst + idx4);
    }
  }

  // Per-feature j-sum across lanes/waves -> pair_to_node[i].
  atomicAdd(&psum[m16],      sum0);
  atomicAdd(&psum[16 + m16], sum1);
  __syncthreads();
  if (tid < ED) p2n[i * ED + tid] = psum[tid] * (1.f / (float)L);
}

// ---------------------------------------------------------------------------
// Kernel C: node = node + [node, pair_to_node] @ W_node + b_node, then LayerNorm.
// ---------------------------------------------------------------------------
__global__ void evo_node(const float* __restrict__ node,
                         const float* __restrict__ p2n,
                         const float* __restrict__ W_node,
                         const float* __restrict__ b_node,
                         const float* __restrict__ ln_scale,
                         const float* __restrict__ ln_offset,
                         float* __restrict__ node_out) {
  __shared__ float nin[ND + ED];
  __shared__ float red[ND];
  __shared__ float red2[ND];
  const int i = blockIdx.x;
  const int t = threadIdx.x;   // 0..63
  nin[t] = node[i * ND + t];
  if (t < ED) nin[ND + t] = p2n[i * ED + t];
  __syncthreads();

  float acc = nin[t] + b_node[t];
#pragma unroll 8
  for (int k = 0; k < ND + ED; ++k)
    acc = fmaf(nin[k], W_node[k * ND + t], acc);

  red[t]  = acc;
  red2[t] = acc * acc;
  __syncthreads();
  for (int s = 32; s > 0; s >>= 1) {
    if (t < s) { red[t] += red[t + s]; red2[t] += red2[t + s]; }
    __syncthreads();
  }
  const float mean = red[0] * (1.f / (float)ND);
  const float var  = red2[0] * (1.f / (float)ND) - mean * mean;
  const float r    = rsqrtf(var + LN_EPS);
  node_out[i * ND + t] = (acc - mean) * r * ln_scale[t] + ln_offset[t];
}

// ---------------------------------------------------------------------------
extern "C" void kernel_launch(void* const* d_in, const int* in_sizes, int n_in,
                              void* d_out, int out_size, void* d_ws, size_t ws_size,
                              hipStream_t stream) {
  (void)in_sizes; (void)n_in; (void)out_size; (void)ws_size;
  const float* node      = (const float*)d_in[0];
  const float* pairf     = (const float*)d_in[1];
  const float* W_pair    = (const float*)d_in[2];
  const float* b_pair    = (const float*)d_in[3];
  const float* W_node    = (const float*)d_in[4];
  const float* b_node    = (const float*)d_in[5];
  const float* ln_scale  = (const float*)d_in[6];
  const float* ln_offset = (const float*)d_in[7];

  float* out_node = (float*)d_out;                   // (1024, 64)
  float* out_pair = out_node + (size_t)L * ND;       // (1024, 1024, 32)

  float* ni  = (float*)d_ws;                         // (1024, 32)
  float* nj  = ni + (size_t)L * ED;                  // (1024, 32)
  float* p2n = nj + (size_t)L * ED;                  // (1024, 32)

  evo_ninj<<<L, ND, 0, stream>>>(node, W_pair, ni, nj);
  evo_pair<<<L, 32 * WAVES, 0, stream>>>(pairf, W_pair, b_pair, ni, nj, out_pair, p2n);
  evo_node<<<L, ND, 0, stream>>>(node, p2n, W_node, b_node, ln_scale, ln_offset, out_node);
}